// NetGAT_6098853560843
// MI455X (gfx1250) — compile-verified
//
#include <hip/hip_runtime.h>
#include <hip/hip_bf16.h>

// ---------------------------------------------------------------------------
// NetGAT for MI455X (gfx1250): fp32 WMMA GEMMs (16x16x4), async global->LDS
// tile staging (ASYNCcnt), atomic segment softmax/max for the graph ops.
// ---------------------------------------------------------------------------

typedef float v2f __attribute__((ext_vector_type(2)));
typedef float v8f __attribute__((ext_vector_type(8)));
typedef __attribute__((address_space(3))) float lds_f;

#define N_NODES 50000
#define N_EDGES 800000
#define DIMK    128
#define DIMC    32
#define NGRAPH  512
#define NEG_SLOPE 0.2f
#define BN_EPS  1e-5f
#define NEG_INF_KEY 0x007FFFFFu

__device__ __forceinline__ float lrelu(float v) { return v > 0.f ? v : NEG_SLOPE * v; }

// order-preserving float -> uint mapping (for atomicMax-based segment max)
__device__ __forceinline__ unsigned fkey(float f) {
    unsigned b = __float_as_uint(f);
    return (b & 0x80000000u) ? ~b : (b | 0x80000000u);
}
__device__ __forceinline__ float funkey(unsigned k) {
    unsigned b = (k & 0x80000000u) ? (k ^ 0x80000000u) : ~k;
    return __uint_as_float(b);
}

// ---------------------------------------------------------------------------
// out[M, OCOLS] = A[M,128] @ W[OCOLS,128]^T (+ bias)   -- fp32 WMMA 16x16x4
// One 16-row strip per block; one 16x16 output tile per wave.
// A tile staged in LDS via GLOBAL_LOAD_ASYNC_TO_LDS_B128 (ASYNCcnt path);
// padded stride 132 words -> conflict free on 64 banks.
// ---------------------------------------------------------------------------
template <int OCOLS>
__global__ __launch_bounds__(32 * (OCOLS / 16))
void gemm_nt_wmma_kernel(const float* __restrict__ A,
                         const float* __restrict__ W,
                         const float* __restrict__ bias,
                         float* __restrict__ out, int M) {
    constexpr int WAVES = OCOLS / 16;
    __shared__ float As[16 * 132];

    const int tid  = threadIdx.x;
    const int row0 = blockIdx.x * 16;
    const bool full_tile = (row0 + 16 <= M);

    if (full_tile) {
        // async DMA: 512 x 16B chunks (16 rows x 32 float4), straight to LDS
        for (int idx = tid; idx < 16 * 32; idx += 32 * WAVES) {
            int r  = idx >> 5;       // row in tile
            int kc = idx & 31;       // float4 chunk within row
            const float* gp = A + (size_t)(row0 + r) * DIMK + kc * 4;
            lds_f* lp = (lds_f*)&As[r * 132 + kc * 4];
            asm volatile("global_load_async_to_lds_b128 %0, %1, off"
                         :: "v"(lp), "v"(gp) : "memory");
        }
        asm volatile("s_wait_asynccnt 0x0" ::: "memory");
    } else {
        for (int idx = tid; idx < 16 * DIMK; idx += 32 * WAVES) {
            int r = idx >> 7, k = idx & 127;
            int gr = row0 + r;
            As[r * 132 + k] = (gr < M) ? A[(size_t)gr * DIMK + k] : 0.0f;
        }
    }
    __syncthreads();

    const int wave  = tid >> 5;
    const int lane  = tid & 31;
    const int c0    = wave * 16;
    const int mrow  = lane & 15;        // A-frag row / D-frag column index
    const int khalf = (lane >> 4) * 2;  // 0 or 2 (K sub-pair per half-wave)

    v8f acc = {};
    const float* Wc = W + (size_t)(c0 + mrow) * DIMK;  // B column = W row

#pragma unroll 4
    for (int k0 = 0; k0 < DIMK; k0 += 4) {
        v2f a, b;
        a.x = As[mrow * 132 + k0 + khalf];
        a.y = As[mrow * 132 + k0 + khalf + 1];
        b.x = Wc[k0 + khalf];
        b.y = Wc[k0 + khalf + 1];
        acc = __builtin_amdgcn_wmma_f32_16x16x4_f32(false, a, false, b,
                                                    (short)0, acc, false, false);
    }

    const int ncol  = c0 + (lane & 15);
    const int rbase = (lane >> 4) * 8;
    const float bv  = bias ? bias[ncol] : 0.0f;
    if (full_tile) {
#pragma unroll
        for (int j = 0; j < 8; ++j)
            out[(size_t)(row0 + rbase + j) * OCOLS + ncol] = acc[j] + bv;
    } else {
#pragma unroll
        for (int j = 0; j < 8; ++j) {
            int gr = row0 + rbase + j;
            if (gr < M) out[(size_t)gr * OCOLS + ncol] = acc[j] + bv;
        }
    }
}

// --------------------------- batch norm helpers ----------------------------
__global__ void fill_f32_kernel(float* p, float v, int n) {
    int i = blockIdx.x * blockDim.x + threadIdx.x;
    if (i < n) p[i] = v;
}
__global__ void fill_u32_kernel(unsigned* p, unsigned v, int n) {
    int i = blockIdx.x * blockDim.x + threadIdx.x;
    if (i < n) p[i] = v;
}

__global__ __launch_bounds__(128)
void colstats_kernel(const float* __restrict__ h, float* __restrict__ sums,
                     float* __restrict__ sumsq, int n) {
    int c = threadIdx.x;                 // 128 columns
    int r0 = blockIdx.x * 64;
    int r1 = r0 + 64 < n ? r0 + 64 : n;
    float s = 0.f, q = 0.f;
    for (int r = r0; r < r1; ++r) {
        float v = h[(size_t)r * DIMK + c];
        s += v; q += v * v;
    }
    atomicAdd(&sums[c], s);
    atomicAdd(&sumsq[c], q);
}

__global__ __launch_bounds__(128)
void bn_params_kernel(const float* __restrict__ stats, const float* __restrict__ g,
                      const float* __restrict__ be, float* __restrict__ scale,
                      float* __restrict__ shift, float inv_n) {
    int c = threadIdx.x;
    float mu  = stats[c] * inv_n;
    float var = stats[128 + c] * inv_n - mu * mu;
    float sc  = g[c] * rsqrtf(var + BN_EPS);
    scale[c] = sc;
    shift[c] = be[c] - mu * sc;
}

__global__ void bn_relu_kernel(float* __restrict__ h, const float* __restrict__ scale,
                               const float* __restrict__ shift, int total) {
    int idx = blockIdx.x * blockDim.x + threadIdx.x;
    if (idx >= total) return;
    int c = idx & 127;
    float v = h[idx] * scale[c] + shift[c];
    h[idx] = v > 0.f ? v : 0.f;
}

// --------------------------- GAT attention ---------------------------------
// one wave32 per node: a_s[n] = hp[n].a_src ; a_d[n] = hp[n].a_dst
__global__ __launch_bounds__(256)
void attn_dots_kernel(const float* __restrict__ hp, const float* __restrict__ a_src,
                      const float* __restrict__ a_dst, float* __restrict__ as_out,
                      float* __restrict__ ad_out, int n) {
    int node = (blockIdx.x * blockDim.x + threadIdx.x) >> 5;
    int lane = threadIdx.x & 31;
    if (node >= n) return;
    float4 h4 = ((const float4*)(hp + (size_t)node * DIMK))[lane];
    float4 s4 = ((const float4*)a_src)[lane];
    float4 d4 = ((const float4*)a_dst)[lane];
    float s = h4.x * s4.x + h4.y * s4.y + h4.z * s4.z + h4.w * s4.w;
    float d = h4.x * d4.x + h4.y * d4.y + h4.z * d4.z + h4.w * d4.w;
#pragma unroll
    for (int off = 16; off; off >>= 1) {
        s += __shfl_down(s, off, 32);
        d += __shfl_down(d, off, 32);
    }
    if (lane == 0) { as_out[node] = s; ad_out[node] = d; }
}

// self-loop logit; seed the per-node segment max with it (always finite)
__global__ void eself_kernel(const float* __restrict__ a_s, const float* __restrict__ a_d,
                             float* __restrict__ e_self, unsigned* __restrict__ m_u, int n) {
    int i = blockIdx.x * blockDim.x + threadIdx.x;
    if (i >= n) return;
    float es = lrelu(a_s[i] + a_d[i]);
    e_self[i] = es;
    m_u[i] = fkey(es);
}

__global__ void edge_max_kernel(const int* __restrict__ src, const int* __restrict__ dst,
                                const int* __restrict__ ea, const float* __restrict__ a_s,
                                const float* __restrict__ a_d, unsigned* __restrict__ m_u,
                                int ne) {
    int e = blockIdx.x * blockDim.x + threadIdx.x;
    if (e >= ne || ea[e] != 1) return;
    float l = lrelu(a_s[src[e]] + a_d[dst[e]]);
    atomicMax(&m_u[dst[e]], fkey(l));
}

__global__ void mdecode_kernel(const unsigned* __restrict__ m_u, const float* __restrict__ e_self,
                               float* __restrict__ m, float* __restrict__ denom, int n) {
    int i = blockIdx.x * blockDim.x + threadIdx.x;
    if (i >= n) return;
    float mv = funkey(m_u[i]);
    m[i] = mv;
    denom[i] = expf(e_self[i] - mv);  // self-loop term
}

__global__ void edge_denom_kernel(const int* __restrict__ src, const int* __restrict__ dst,
                                  const int* __restrict__ ea, const float* __restrict__ a_s,
                                  const float* __restrict__ a_d, const float* __restrict__ m,
                                  float* __restrict__ denom, int ne) {
    int e = blockIdx.x * blockDim.x + threadIdx.x;
    if (e >= ne || ea[e] != 1) return;
    int d = dst[e];
    float l = lrelu(a_s[src[e]] + a_d[d]);
    atomicAdd(&denom[d], expf(l - m[d]));
}

// h_out[n,:] = bias + alpha_self * hp[n,:]
__global__ void gat_self_init_kernel(const float* __restrict__ hp, const float* __restrict__ e_self,
                                     const float* __restrict__ m, const float* __restrict__ denom,
                                     const float* __restrict__ bias, float* __restrict__ hout,
                                     int total) {
    int idx = blockIdx.x * blockDim.x + threadIdx.x;
    if (idx >= total) return;
    int node = idx >> 7, c = idx & 127;
    float sw = expf(e_self[node] - m[node]) / denom[node];
    hout[idx] = bias[c] + sw * hp[idx];
}

// one wave32 per edge: hout[dst] += alpha_e * hp[src]
__global__ __launch_bounds__(256)
void edge_message_kernel(const int* __restrict__ src, const int* __restrict__ dst,
                         const int* __restrict__ ea, const float* __restrict__ a_s,
                         const float* __restrict__ a_d, const float* __restrict__ m,
                         const float* __restrict__ denom, const float* __restrict__ hp,
                         float* __restrict__ hout, int ne) {
    int e = (blockIdx.x * blockDim.x + threadIdx.x) >> 5;
    int lane = threadIdx.x & 31;
    if (e >= ne || ea[e] != 1) return;
    int s = src[e], d = dst[e];
    float w = 0.f;
    if (lane == 0) {
        float l = lrelu(a_s[s] + a_d[d]);
        w = expf(l - m[d]) / denom[d];
    }
    w = __shfl(w, 0, 32);
    float4 hv = ((const float4*)(hp + (size_t)s * DIMK))[lane];
    float* o = hout + (size_t)d * DIMK + lane * 4;
    atomicAdd(o + 0, w * hv.x);
    atomicAdd(o + 1, w * hv.y);
    atomicAdd(o + 2, w * hv.z);
    atomicAdd(o + 3, w * hv.w);
}

// --------------------------- pooling / output ------------------------------
template <int DIM>
__global__ void pool_max_kernel(const float* __restrict__ h, const int* __restrict__ batch,
                                unsigned* __restrict__ pooled_u, int n) {
    int idx = blockIdx.x * blockDim.x + threadIdx.x;
    if (idx >= n * DIM) return;
    int node = idx / DIM, c = idx & (DIM - 1);
    atomicMax(&pooled_u[(size_t)batch[node] * DIM + c], fkey(h[idx]));
}

__global__ void pool_decode_kernel(const unsigned* __restrict__ pu, float* __restrict__ p, int n) {
    int i = blockIdx.x * blockDim.x + threadIdx.x;
    if (i < n) p[i] = funkey(pu[i]);
}

// out[g,c] += pooled[g,:] . W[c,:] + b[c]
__global__ void project_add_kernel(const float* __restrict__ pooled, const float* __restrict__ W,
                                   const float* __restrict__ b, float* __restrict__ out) {
    int idx = blockIdx.x * blockDim.x + threadIdx.x;
    if (idx >= NGRAPH * DIMC) return;
    int g = idx >> 5, c = idx & 31;
    const float4* pg = (const float4*)(pooled + (size_t)g * DIMK);
    const float4* wc = (const float4*)(W + (size_t)c * DIMK);
    float acc = b[c];
#pragma unroll
    for (int k = 0; k < 32; ++k) {
        float4 a = pg[k], w = wc[k];
        acc += a.x * w.x + a.y * w.y + a.z * w.z + a.w * w.w;
    }
    out[idx] += acc;
}

// ---------------------------------------------------------------------------
extern "C" void kernel_launch(void* const* d_in, const int* in_sizes, int n_in,
                              void* d_out, int out_size, void* d_ws, size_t ws_size,
                              hipStream_t stream) {
    (void)in_sizes; (void)n_in; (void)out_size; (void)ws_size;
    // inputs in setup_inputs() insertion order
    const float* x     = (const float*)d_in[0];
    const int*   esrc  = (const int*)d_in[1];              // edge_index[0]
    const int*   edst  = ((const int*)d_in[1]) + N_EDGES;  // edge_index[1]
    const int*   eattr = (const int*)d_in[2];
    const int*   batch = (const int*)d_in[3];
    const float* w1  = (const float*)d_in[4];
    const float* b1  = (const float*)d_in[5];
    const float* g1  = (const float*)d_in[6];
    const float* be1 = (const float*)d_in[7];
    const float* w2  = (const float*)d_in[8];
    const float* b2  = (const float*)d_in[9];
    const float* g2  = (const float*)d_in[10];
    const float* be2 = (const float*)d_in[11];
    const float* wl0 = (const float*)d_in[12];
    const float* bl0 = (const float*)d_in[13];
    const float* gatW[2]    = {(const float*)d_in[14], (const float*)d_in[18]};
    const float* gatAsrc[2] = {(const float*)d_in[15], (const float*)d_in[19]};
    const float* gatAdst[2] = {(const float*)d_in[16], (const float*)d_in[20]};
    const float* gatBias[2] = {(const float*)d_in[17], (const float*)d_in[21]};
    const float* linW[2]    = {(const float*)d_in[22], (const float*)d_in[24]};
    const float* linB[2]    = {(const float*)d_in[23], (const float*)d_in[25]};
    float* out = (float*)d_out;

    // workspace layout (fp32 units)
    float* ws     = (float*)d_ws;
    float* h_a    = ws;                       // [N,128]
    float* h_b    = h_a + (size_t)N_NODES * DIMK;
    float* hp     = h_b + (size_t)N_NODES * DIMK;   // also used as [N,32] temp
    float* a_s    = hp + (size_t)N_NODES * DIMK;
    float* a_d    = a_s + N_NODES;
    float* e_self = a_d + N_NODES;
    float* mvals  = e_self + N_NODES;
    float* denom  = mvals + N_NODES;
    unsigned* m_u = (unsigned*)(denom + N_NODES);
    float* stats  = (float*)(m_u + N_NODES);  // sums[128] sumsq[128]
    float* scale  = stats + 256;              // [128]
    float* shift  = scale + 128;              // [128]
    unsigned* pooled_u = (unsigned*)(shift + 128);      // [G,128] keys
    float* pooled      = (float*)(pooled_u + NGRAPH * DIMK);

    const int TB = 256;
    auto blocks = [](long total, int tb) { return (unsigned)((total + tb - 1) / tb); };
    const unsigned gemm_blocks = (N_NODES + 15) / 16;
    const long totND = (long)N_NODES * DIMK;

    // ---- Stage A: h_a = relu(BN(x @ w1^T + b1)) ----
    gemm_nt_wmma_kernel<128><<<gemm_blocks, 256, 0, stream>>>(x, w1, b1, h_a, N_NODES);
    fill_f32_kernel<<<1, 256, 0, stream>>>(stats, 0.f, 256);
    colstats_kernel<<<blocks(N_NODES, 64), 128, 0, stream>>>(h_a, stats, stats + 128, N_NODES);
    bn_params_kernel<<<1, 128, 0, stream>>>(stats, g1, be1, scale, shift, 1.0f / N_NODES);
    bn_relu_kernel<<<blocks(totND, TB), TB, 0, stream>>>(h_a, scale, shift, (int)totND);

    // ---- Stage B: h_b = relu(BN(h_a @ w2^T + b2)) ----
    gemm_nt_wmma_kernel<128><<<gemm_blocks, 256, 0, stream>>>(h_a, w2, b2, h_b, N_NODES);
    fill_f32_kernel<<<1, 256, 0, stream>>>(stats, 0.f, 256);
    colstats_kernel<<<blocks(N_NODES, 64), 128, 0, stream>>>(h_b, stats, stats + 128, N_NODES);
    bn_params_kernel<<<1, 128, 0, stream>>>(stats, g2, be2, scale, shift, 1.0f / N_NODES);
    bn_relu_kernel<<<blocks(totND, TB), TB, 0, stream>>>(h_b, scale, shift, (int)totND);

    // ---- Stage C: out = segment_max(h_b @ wl0^T + bl0) ----
    gemm_nt_wmma_kernel<32><<<gemm_blocks, 64, 0, stream>>>(h_b, wl0, bl0, hp, N_NODES);
    fill_u32_kernel<<<blocks(NGRAPH * DIMC, TB), TB, 0, stream>>>(pooled_u, NEG_INF_KEY, NGRAPH * DIMC);
    pool_max_kernel<DIMC><<<blocks((long)N_NODES * DIMC, TB), TB, 0, stream>>>(hp, batch, pooled_u, N_NODES);
    pool_decode_kernel<<<blocks(NGRAPH * DIMC, TB), TB, 0, stream>>>(pooled_u, out, NGRAPH * DIMC);

    // ---- GAT layers ----
    const float* hin = h_b;
    float* hout = h_a;
    for (int i = 0; i < 2; ++i) {
        gemm_nt_wmma_kernel<128><<<gemm_blocks, 256, 0, stream>>>(hin, gatW[i], nullptr, hp, N_NODES);
        attn_dots_kernel<<<blocks((long)N_NODES * 32, TB), TB, 0, stream>>>(hp, gatAsrc[i], gatAdst[i], a_s, a_d, N_NODES);
        eself_kernel<<<blocks(N_NODES, TB), TB, 0, stream>>>(a_s, a_d, e_self, m_u, N_NODES);
        edge_max_kernel<<<blocks(N_EDGES, TB), TB, 0, stream>>>(esrc, edst, eattr, a_s, a_d, m_u, N_EDGES);
        mdecode_kernel<<<blocks(N_NODES, TB), TB, 0, stream>>>(m_u, e_self, mvals, denom, N_NODES);
        edge_denom_kernel<<<blocks(N_EDGES, TB), TB, 0, stream>>>(esrc, edst, eattr, a_s, a_d, mvals, denom, N_EDGES);
        gat_self_init_kernel<<<blocks(totND, TB), TB, 0, stream>>>(hp, e_self, mvals, denom, gatBias[i], hout, (int)totND);
        edge_message_kernel<<<blocks((long)N_EDGES * 32, TB), TB, 0, stream>>>(esrc, edst, eattr, a_s, a_d, mvals, denom, hp, hout, N_EDGES);

        fill_u32_kernel<<<blocks(NGRAPH * DIMK, TB), TB, 0, stream>>>(pooled_u, NEG_INF_KEY, NGRAPH * DIMK);
        pool_max_kernel<DIMK><<<blocks(totND, TB), TB, 0, stream>>>(hout, batch, pooled_u, N_NODES);
        pool_decode_kernel<<<blocks(NGRAPH * DIMK, TB), TB, 0, stream>>>(pooled_u, pooled, NGRAPH * DIMK);
        project_add_kernel<<<blocks(NGRAPH * DIMC, TB), TB, 0, stream>>>(pooled, linW[i], linB[i], out);

        const float* t = hin; hin = hout; hout = (float*)t;
    }
}